// StatAnalModule_60215441490087
// MI455X (gfx1250) — compile-verified
//
#include <hip/hip_runtime.h>

#ifndef __has_builtin
#define __has_builtin(x) 0
#endif

#if __has_builtin(__builtin_amdgcn_global_load_async_to_lds_b128) && \
    __has_builtin(__builtin_amdgcn_s_wait_asynccnt)
#define USE_ASYNC_LDS 1
#else
#define USE_ASYNC_LDS 0
#endif

typedef float v4f __attribute__((ext_vector_type(4)));
typedef int   v4i __attribute__((ext_vector_type(4)));

static constexpr int kB      = 64;
static constexpr int kC      = 8;
static constexpr int kL      = 131072;
static constexpr int kTPB    = 512;             // 16 waves (wave32)
static constexpr int kWaves  = kTPB / 32;
static constexpr int kChunk  = kTPB * 4;        // floats per block-chunk (8 KB)
static constexpr int kNChunk = kL / kChunk;     // 64
static constexpr int kSlots  = 4;               // LDS ring slots
static constexpr int kAhead  = 3;               // async chunks in flight per wave

// ---------------- wave32 pairwise reductions ----------------
__device__ __forceinline__ float wred_sum(float v) {
#pragma unroll
  for (int m = 16; m > 0; m >>= 1) v += __shfl_xor(v, m, 32);
  return v;
}
__device__ __forceinline__ float wred_max(float v) {
#pragma unroll
  for (int m = 16; m > 0; m >>= 1) v = fmaxf(v, __shfl_xor(v, m, 32));
  return v;
}
__device__ __forceinline__ float wred_min(float v) {
#pragma unroll
  for (int m = 16; m > 0; m >>= 1) v = fminf(v, __shfl_xor(v, m, 32));
  return v;
}

// ---------------- per-element accumulation (~7 VALU) ----------------
__device__ __forceinline__ void acc1(float f, float& s1, float& s2, float& s3,
                                     float& s4, float& sa, float& mx, float& mn) {
  s1 += f;
  float f2 = f * f;
  s2 = fmaf(f, f, s2);
  s3 = fmaf(f2, f, s3);
  s4 = fmaf(f2, f2, s4);
  sa += fabsf(f);
  mx = fmaxf(mx, f);
  mn = fminf(mn, f);
}
__device__ __forceinline__ void acc4(v4f v, float& s1, float& s2, float& s3,
                                     float& s4, float& sa, float& mx, float& mn) {
  acc1(v.x, s1, s2, s3, s4, sa, mx, mn);
  acc1(v.y, s1, s2, s3, s4, sa, mx, mn);
  acc1(v.z, s1, s2, s3, s4, sa, mx, mn);
  acc1(v.w, s1, s2, s3, s4, sa, mx, mn);
}

#if USE_ASYNC_LDS
__device__ __forceinline__ void async_load16(const float* g, v4f* l) {
  __builtin_amdgcn_global_load_async_to_lds_b128(
      (__attribute__((address_space(1))) v4i*)g,
      (__attribute__((address_space(3))) v4i*)l,
      /*offset=*/0, /*cpol=*/0);
}
// s_wait_asynccnt takes an immediate -> constant-folded switch
__device__ __forceinline__ void wait_async_le(int n) {
  switch (n) {
    case 0: __builtin_amdgcn_s_wait_asynccnt(0); break;
    case 1: __builtin_amdgcn_s_wait_asynccnt(1); break;
    default: __builtin_amdgcn_s_wait_asynccnt(2); break;
  }
}
#endif

__global__ __launch_bounds__(kTPB) void stat_kernel(const float* __restrict__ x,
                                                    float* __restrict__ out) {
  const int row = blockIdx.x;            // 0 .. B*C-1
  const int tid = threadIdx.x;
  const float* xr = x + (size_t)row * kL;

  float s1 = 0.f, s2 = 0.f, s3 = 0.f, s4 = 0.f, sa = 0.f;
  float mx = -__builtin_inff(), mn = __builtin_inff();

#if USE_ASYNC_LDS
  __shared__ v4f buf[kSlots][kTPB];      // 32 KB ring; each lane touches only its slot
  // prologue: kAhead chunks in flight
#pragma unroll
  for (int c = 0; c < kAhead; ++c)
    async_load16(xr + (size_t)c * kChunk + tid * 4, &buf[c & (kSlots - 1)][tid]);

  for (int c = 0; c < kNChunk; ++c) {
    int newer = kNChunk - 1 - c;
    if (newer > kAhead - 1) newer = kAhead - 1;
    wait_async_le(newer);                           // chunk c has landed in LDS
    v4f v = buf[c & (kSlots - 1)][tid];             // ds_load_b128 (own bytes only)
    acc4(v, s1, s2, s3, s4, sa, mx, mn);
    int nx = c + kAhead;                            // refill a slot consumed last iter
    if (nx < kNChunk)
      async_load16(xr + (size_t)nx * kChunk + tid * 4, &buf[nx & (kSlots - 1)][tid]);
  }
#else
  const v4f* xv = (const v4f*)xr;
#pragma unroll 4
  for (int c = 0; c < kNChunk; ++c) {
    if (c + 8 < kNChunk) __builtin_prefetch(&xv[(size_t)(c + 8) * kTPB + tid], 0, 0);
    v4f v = __builtin_nontemporal_load(&xv[(size_t)c * kTPB + tid]);
    acc4(v, s1, s2, s3, s4, sa, mx, mn);
  }
#endif

  // ---- intra-wave pairwise tree ----
  s1 = wred_sum(s1); s2 = wred_sum(s2); s3 = wred_sum(s3);
  s4 = wred_sum(s4); sa = wred_sum(sa);
  mx = wred_max(mx); mn = wred_min(mn);

  __shared__ float part[7][kWaves];
  const int wid = tid >> 5, lane = tid & 31;
  if (lane == 0) {
    part[0][wid] = s1; part[1][wid] = s2; part[2][wid] = s3;
    part[3][wid] = s4; part[4][wid] = sa; part[5][wid] = mx; part[6][wid] = mn;
  }
  __syncthreads();

  if (tid < 32) {
    const bool ok = tid < kWaves;
    float a1 = ok ? part[0][tid] : 0.f;
    float a2 = ok ? part[1][tid] : 0.f;
    float a3 = ok ? part[2][tid] : 0.f;
    float a4 = ok ? part[3][tid] : 0.f;
    float a5 = ok ? part[4][tid] : 0.f;
    float amx = ok ? part[5][tid] : -__builtin_inff();
    float amn = ok ? part[6][tid] : __builtin_inff();
    a1 = wred_sum(a1); a2 = wred_sum(a2); a3 = wred_sum(a3);
    a4 = wred_sum(a4); a5 = wred_sum(a5);
    amx = wred_max(amx); amn = wred_min(amn);

    if (tid == 0) {
      const double Lf = (double)kL;
      const double S1 = a1, S2 = a2, S3 = a3, S4 = a4, SA = a5;
      const double MX = amx, MN = amn;
      const double mean = S1 / Lf;
      const double rms  = sqrt(S2 / Lf);
      const double var  = (S2 - Lf * mean * mean) / (Lf - 1.0);
      const double sd   = sqrt(var);
      const double m3   = S3 - 3.0 * mean * S2 + 2.0 * Lf * mean * mean * mean;
      const double m4   = S4 - 4.0 * mean * S3 + 6.0 * mean * mean * S2 -
                          3.0 * Lf * mean * mean * mean * mean;
      const double skew = m3 / (sd * sd * sd) * Lf / ((Lf - 1.0) * (Lf - 2.0));
      const double kurt = Lf / ((Lf - 1.0) * (Lf - 1.0)) * m4 / (var * var) - 3.0;
      const double absmax  = fmax(MX, -MN);
      const double meanabs = SA / Lf;
      const double crest   = absmax / rms;
      const double impulse = (MX - MN) / meanabs;
      const double shape   = rms / meanabs;

      const int b = row / kC, c = row % kC;
      float* o = out + (size_t)b * (10 * kC);
      o[0 * kC + c] = (float)mean;
      o[1 * kC + c] = (float)MX;
      o[2 * kC + c] = (float)MN;
      o[3 * kC + c] = (float)rms;
      o[4 * kC + c] = (float)var;
      o[5 * kC + c] = (float)skew;
      o[6 * kC + c] = (float)kurt;
      o[7 * kC + c] = (float)crest;
      o[8 * kC + c] = (float)impulse;
      o[9 * kC + c] = (float)shape;
    }
  }
}

extern "C" void kernel_launch(void* const* d_in, const int* in_sizes, int n_in,
                              void* d_out, int out_size, void* d_ws, size_t ws_size,
                              hipStream_t stream) {
  const float* x = (const float*)d_in[0];
  float* out = (float*)d_out;
  stat_kernel<<<kB * kC, kTPB, 0, stream>>>(x, out);
  (void)in_sizes; (void)n_in; (void)out_size; (void)d_ws; (void)ws_size;
}